// GCN_30391188586834
// MI455X (gfx1250) — compile-verified
//
#include <hip/hip_runtime.h>

typedef __attribute__((ext_vector_type(2))) float v2f;
typedef __attribute__((ext_vector_type(8))) float v8f;

#define F_IN  256
#define F_HID 64
#define F_OUT 32

// ---------------------------------------------------------------------------
// Degrees: deg_out[src[e]] += 1 ; deg_in[dst[e]] += 1
// ---------------------------------------------------------------------------
__global__ void gcn_degrees(const int* __restrict__ src, const int* __restrict__ dst,
                            float* __restrict__ deg_out, float* __restrict__ deg_in, int nE) {
    int e = blockIdx.x * blockDim.x + threadIdx.x;
    if (e < nE) {
        atomicAdd(deg_out + src[e], 1.0f);
        atomicAdd(deg_in  + dst[e], 1.0f);
    }
}

// In-place: deg -> rsqrt(max(deg,1))
__global__ void gcn_norms(float* __restrict__ deg_out, float* __restrict__ deg_in, int n) {
    int i = blockIdx.x * blockDim.x + threadIdx.x;
    if (i < n) {
        deg_out[i] = rsqrtf(fmaxf(deg_out[i], 1.0f));
        deg_in[i]  = rsqrtf(fmaxf(deg_in[i],  1.0f));
    }
}

// ---------------------------------------------------------------------------
// GEMM1: H[n,64] = (X[n,256] * norm_out[n]) @ W[256,64]   (fp32 WMMA 16x16x4)
// One wave per 16-row tile; 4 column tiles of 16.
// ---------------------------------------------------------------------------
__global__ void gcn_gemm1(const float* __restrict__ X, const float* __restrict__ W,
                          const float* __restrict__ norm_out, float* __restrict__ H,
                          int ntiles) {
    int wave = (int)((blockIdx.x * blockDim.x + threadIdx.x) >> 5);
    if (wave >= ntiles) return;                 // wave-uniform: EXEC stays all-ones
    int lane = threadIdx.x & 31;
    int m    = lane & 15;
    int half = lane >> 4;
    size_t row = (size_t)wave * 16 + m;
    float  no  = norm_out[row];
    const float* xrow = X + row * F_IN;

    v8f acc[4];
    #pragma unroll
    for (int t = 0; t < 4; ++t) acc[t] = (v8f){};

    #pragma unroll 4
    for (int k = 0; k < F_IN; k += 4) {
        float2 a2 = *(const float2*)(xrow + k + 2 * half);
        v2f a; a.x = a2.x * no; a.y = a2.y * no;
        const float* wk0 = W + (size_t)(k + 2 * half) * F_HID;
        #pragma unroll
        for (int t = 0; t < 4; ++t) {
            v2f b;
            b.x = wk0[t * 16 + m];
            b.y = wk0[F_HID + t * 16 + m];
            acc[t] = __builtin_amdgcn_wmma_f32_16x16x4_f32(
                false, a, false, b, (short)0, acc[t], false, false);
        }
    }

    #pragma unroll
    for (int t = 0; t < 4; ++t) {
        #pragma unroll
        for (int r = 0; r < 8; ++r) {
            H[((size_t)wave * 16 + r + 8 * half) * F_HID + t * 16 + m] = acc[t][r];
        }
    }
}

// ---------------------------------------------------------------------------
// GEMM2: H[n,32] = ((AGG[n,64]*norm_in[n] + b1) * norm_out[n]) @ W[64,32]
// ---------------------------------------------------------------------------
__global__ void gcn_gemm2(const float* __restrict__ AGG, const float* __restrict__ W,
                          const float* __restrict__ b1, const float* __restrict__ norm_in,
                          const float* __restrict__ norm_out, float* __restrict__ H,
                          int ntiles) {
    int wave = (int)((blockIdx.x * blockDim.x + threadIdx.x) >> 5);
    if (wave >= ntiles) return;
    int lane = threadIdx.x & 31;
    int m    = lane & 15;
    int half = lane >> 4;
    size_t row = (size_t)wave * 16 + m;
    float ni = norm_in[row];
    float no = norm_out[row];
    const float* arow = AGG + row * F_HID;

    v8f acc[2];
    acc[0] = (v8f){}; acc[1] = (v8f){};

    #pragma unroll
    for (int k = 0; k < F_HID; k += 4) {
        float2 a2 = *(const float2*)(arow + k + 2 * half);
        float2 bb = *(const float2*)(b1 + k + 2 * half);
        v2f a;
        a.x = (a2.x * ni + bb.x) * no;
        a.y = (a2.y * ni + bb.y) * no;
        const float* wk0 = W + (size_t)(k + 2 * half) * F_OUT;
        #pragma unroll
        for (int t = 0; t < 2; ++t) {
            v2f b;
            b.x = wk0[t * 16 + m];
            b.y = wk0[F_OUT + t * 16 + m];
            acc[t] = __builtin_amdgcn_wmma_f32_16x16x4_f32(
                false, a, false, b, (short)0, acc[t], false, false);
        }
    }

    #pragma unroll
    for (int t = 0; t < 2; ++t) {
        #pragma unroll
        for (int r = 0; r < 8; ++r) {
            H[((size_t)wave * 16 + r + 8 * half) * F_OUT + t * 16 + m] = acc[t][r];
        }
    }
}

// ---------------------------------------------------------------------------
// Scatter: AGG[dst[e], f] += H[src[e], f]   (coalesced per-feature atomics)
// ---------------------------------------------------------------------------
template <int F>
__global__ void gcn_scatter(const float* __restrict__ H, const int* __restrict__ src,
                            const int* __restrict__ dst, float* __restrict__ AGG, int nE) {
    unsigned tid = blockIdx.x * blockDim.x + threadIdx.x;
    unsigned e = tid / F;
    unsigned f = tid % F;
    if (e < (unsigned)nE) {
        float v = H[(size_t)src[e] * F + f];
        atomicAdd(AGG + (size_t)dst[e] * F + f, v);
    }
}

// out[n,c] = agg2[n,c] * norm_in[n] + b2[c]
__global__ void gcn_final(const float* __restrict__ agg2, const float* __restrict__ norm_in,
                          const float* __restrict__ b2, float* __restrict__ out, int total) {
    int i = blockIdx.x * blockDim.x + threadIdx.x;
    if (i < total) {
        int nidx = i >> 5;      // F_OUT == 32
        int c    = i & 31;
        out[i] = agg2[i] * norm_in[nidx] + b2[c];
    }
}

// ---------------------------------------------------------------------------
extern "C" void kernel_launch(void* const* d_in, const int* in_sizes, int n_in,
                              void* d_out, int out_size, void* d_ws, size_t ws_size,
                              hipStream_t stream) {
    const float* feat = (const float*)d_in[0];   // [n, 256]
    const float* W1   = (const float*)d_in[1];   // [256, 64]
    const float* b1   = (const float*)d_in[2];   // [64]
    const float* W2   = (const float*)d_in[3];   // [64, 32]
    const float* b2   = (const float*)d_in[4];   // [32]
    const int*   src  = (const int*)d_in[5];     // [E]
    const int*   dst  = (const int*)d_in[6];     // [E]
    float* out = (float*)d_out;

    const int n = in_sizes[0] / F_IN;            // 50000
    const int e = in_sizes[5];                   // 800000
    const int ntiles = n / 16;                   // 3125 (N divisible by 16)

    float* ws      = (float*)d_ws;
    float* deg_out = ws;                         // n   (becomes norm_out)
    float* deg_in  = deg_out + n;                // n   (becomes norm_in)
    float* h1      = deg_in + n;                 // n*64
    float* agg1    = h1 + (size_t)n * F_HID;     // n*64
    float* h2      = agg1 + (size_t)n * F_HID;   // n*32
    float* agg2    = h2 + (size_t)n * F_OUT;     // n*32

    // zero accumulators (graph-capture-safe stream memsets)
    hipMemsetAsync(deg_out, 0, (size_t)2 * n * sizeof(float), stream);
    hipMemsetAsync(agg1,    0, (size_t)n * F_HID * sizeof(float), stream);
    hipMemsetAsync(agg2,    0, (size_t)n * F_OUT * sizeof(float), stream);

    gcn_degrees<<<(e + 255) / 256, 256, 0, stream>>>(src, dst, deg_out, deg_in, e);
    gcn_norms<<<(n + 255) / 256, 256, 0, stream>>>(deg_out, deg_in, n);

    // layer 1
    gcn_gemm1<<<(ntiles + 3) / 4, 128, 0, stream>>>(feat, W1, deg_out, h1, ntiles);
    {
        long long total = (long long)e * F_HID;
        gcn_scatter<F_HID><<<(unsigned)((total + 255) / 256), 256, 0, stream>>>(h1, src, dst, agg1, e);
    }

    // layer 2 (norm_in*agg + b1, then norm_out, fused into A-fragment load)
    gcn_gemm2<<<(ntiles + 3) / 4, 128, 0, stream>>>(agg1, W2, b1, deg_in, deg_out, h2, ntiles);
    {
        long long total = (long long)e * F_OUT;
        gcn_scatter<F_OUT><<<(unsigned)((total + 255) / 256), 256, 0, stream>>>(h2, src, dst, agg2, e);
    }

    gcn_final<<<(n * F_OUT + 255) / 256, 256, 0, stream>>>(agg2, deg_in, b2, out, n * F_OUT);
}